// TrafficGNN_39127152067222
// MI455X (gfx1250) — compile-verified
//
#include <hip/hip_runtime.h>

#define N_NODES 100000
#define N_EDGES 3200000
#define F_IN    128
#define HID     16
#define NCLS    10
#define TILES   (N_NODES / 16)   // 6250, exact

typedef __attribute__((ext_vector_type(2))) float v2f;
typedef __attribute__((ext_vector_type(8))) float v8f;

// ---------------- utility ----------------
__global__ void zero_kernel(float* __restrict__ p, int n) {
    int i = blockIdx.x * blockDim.x + threadIdx.x;
    if (i < n) p[i] = 0.0f;
}

__global__ void degree_kernel(const int* __restrict__ dst, float* __restrict__ deg) {
    int e = blockIdx.x * blockDim.x + threadIdx.x;
    if (e < N_EDGES) atomicAdd(&deg[dst[e]], 1.0f);
}

__global__ void dinv_kernel(float* __restrict__ deg) {
    int i = blockIdx.x * blockDim.x + threadIdx.x;
    if (i < N_NODES) deg[i] = __frsqrt_rn(deg[i] + 1.0f);   // +1 self-loop; deg>=1 safe
}

// ---------------- GEMM1: h1 = x @ W1  (fp32 WMMA 16x16x4) ----------------
__global__ __launch_bounds__(128) void gemm1_kernel(const float* __restrict__ x,
                                                    const float* __restrict__ W1,
                                                    float* __restrict__ h1) {
    __shared__ float sW[F_IN * HID];        // 8 KB
    __shared__ float sA[4][16 * F_IN];      // 32 KB (one 16x128 tile per wave)
    const int lane = threadIdx.x & 31;
    const int wv   = threadIdx.x >> 5;
    const int tile = blockIdx.x * 4 + wv;
    const bool valid = (tile < TILES);      // wave-uniform

    for (int i = threadIdx.x; i < F_IN * HID; i += 128) sW[i] = W1[i];
    if (valid) {
        const float4* s4 = (const float4*)(x + (size_t)tile * 16 * F_IN);
        float4* d4 = (float4*)sA[wv];
        #pragma unroll 4
        for (int i = lane; i < 16 * F_IN / 4; i += 32) d4[i] = s4[i];
    }
    __syncthreads();

    if (!valid) return;

    const int n    = lane & 15;             // output column
    const int m    = lane & 15;             // A row
    const int koff = (lane >> 4) << 1;      // 0 for lanes 0-15, 2 for lanes 16-31
    v8f acc = {0.f,0.f,0.f,0.f,0.f,0.f,0.f,0.f};

    #pragma unroll
    for (int ks = 0; ks < F_IN / 4; ++ks) {
        const int kb = ks * 4 + koff;
        v2f a, b;
        a.x = sA[wv][m * F_IN + kb];
        a.y = sA[wv][m * F_IN + kb + 1];
        b.x = sW[(kb    ) * HID + n];
        b.y = sW[(kb + 1) * HID + n];
        acc = __builtin_amdgcn_wmma_f32_16x16x4_f32(false, a, false, b,
                                                    (short)0, acc, false, false);
    }
    const int mbase = (lane >> 4) << 3;     // rows 0-7 or 8-15
    #pragma unroll
    for (int r = 0; r < 8; ++r)
        h1[(size_t)(tile * 16 + mbase + r) * HID + n] = acc[r];
}

// ---------------- GEMM2: h2 = relu_h1 @ W2 (W2 zero-padded 10->16 cols) ----------------
__global__ __launch_bounds__(128) void gemm2_kernel(const float* __restrict__ h,
                                                    const float* __restrict__ W2,
                                                    float* __restrict__ h2) {
    const int lane = threadIdx.x & 31;
    const int wv   = threadIdx.x >> 5;
    const int tile = blockIdx.x * 4 + wv;
    if (tile >= TILES) return;              // wave-uniform

    const int n    = lane & 15;
    const int m    = lane & 15;
    const int koff = (lane >> 4) << 1;
    const float* arow = h + (size_t)(tile * 16 + m) * HID;
    v8f acc = {0.f,0.f,0.f,0.f,0.f,0.f,0.f,0.f};

    #pragma unroll
    for (int ks = 0; ks < HID / 4; ++ks) {
        const int kb = ks * 4 + koff;
        v2f a, b;
        a.x = arow[kb];
        a.y = arow[kb + 1];
        b.x = (n < NCLS) ? W2[(kb    ) * NCLS + n] : 0.0f;
        b.y = (n < NCLS) ? W2[(kb + 1) * NCLS + n] : 0.0f;
        acc = __builtin_amdgcn_wmma_f32_16x16x4_f32(false, a, false, b,
                                                    (short)0, acc, false, false);
    }
    // only the 10 real output columns are stored (divergence after WMMA is fine)
    if (n < NCLS) {
        const int mbase = (lane >> 4) << 3;
        #pragma unroll
        for (int r = 0; r < 8; ++r)
            h2[(size_t)(tile * 16 + mbase + r) * HID + n] = acc[r];
    }
}

// ---------------- edge gather + scatter-add (NC = live columns) ----------------
template <int NC>
__global__ void scatter_kernel(const int* __restrict__ src, const int* __restrict__ dst,
                               const float* __restrict__ dinv, const float* __restrict__ h,
                               float* __restrict__ agg) {
    int e = blockIdx.x * blockDim.x + threadIdx.x;
    if (e >= N_EDGES) return;
    const int s = src[e], d = dst[e];
    const float coef = dinv[s] * dinv[d];
    const float* hs = h + (size_t)s * HID;     // rows are 64B aligned
    float* ad = agg + (size_t)d * HID;
    float v[NC];
    #pragma unroll
    for (int c = 0; c < NC; ++c) v[c] = hs[c];     // merges to b128/b64 loads
    #pragma unroll
    for (int c = 0; c < NC; ++c) atomicAdd(&ad[c], v[c] * coef);
}

// ---------------- self-loop + bias + ReLU (in place into agg) ----------------
__global__ void post1_kernel(float* __restrict__ agg, const float* __restrict__ h1,
                             const float* __restrict__ dinv, const float* __restrict__ b1) {
    int i = blockIdx.x * blockDim.x + threadIdx.x;
    if (i >= N_NODES * HID) return;
    const int nd = i >> 4, c = i & 15;
    const float dv = dinv[nd];
    float v = agg[i] + h1[i] * dv * dv + b1[c];
    agg[i] = v > 0.0f ? v : 0.0f;
}

// ---------------- self-loop + bias + softmax over 10 classes ----------------
__global__ void softmax_kernel(const float* __restrict__ agg2, const float* __restrict__ h2,
                               const float* __restrict__ dinv, const float* __restrict__ b2,
                               float* __restrict__ out) {
    int nd = blockIdx.x * blockDim.x + threadIdx.x;
    if (nd >= N_NODES) return;
    const float dv = dinv[nd];
    const float d2 = dv * dv;
    float logit[NCLS];
    float mx = -3.0e38f;
    #pragma unroll
    for (int c = 0; c < NCLS; ++c) {
        float v = agg2[(size_t)nd * HID + c] + h2[(size_t)nd * HID + c] * d2 + b2[c];
        logit[c] = v;
        mx = fmaxf(mx, v);
    }
    float sum = 0.0f;
    #pragma unroll
    for (int c = 0; c < NCLS; ++c) {
        float ev = __expf(logit[c] - mx);
        logit[c] = ev;
        sum += ev;
    }
    const float inv = 1.0f / sum;
    #pragma unroll
    for (int c = 0; c < NCLS; ++c)
        out[(size_t)nd * NCLS + c] = logit[c] * inv;
}

// ---------------- host entry ----------------
extern "C" void kernel_launch(void* const* d_in, const int* in_sizes, int n_in,
                              void* d_out, int out_size, void* d_ws, size_t ws_size,
                              hipStream_t stream) {
    const float* x  = (const float*)d_in[0];
    const int*   ei = (const int*)  d_in[1];
    const float* W1 = (const float*)d_in[2];
    const float* b1 = (const float*)d_in[3];
    const float* W2 = (const float*)d_in[4];
    const float* b2 = (const float*)d_in[5];
    const int* srcI = ei;
    const int* dstI = ei + N_EDGES;

    // workspace layout (floats): [deg|agg1|agg2|h1|h2]; first 3 zeroed each call
    float* ws   = (float*)d_ws;
    float* deg  = ws;                               // N          (becomes dinv)
    float* agg1 = deg  + N_NODES;                   // N*16
    float* agg2 = agg1 + (size_t)N_NODES * HID;     // N*16
    float* h1   = agg2 + (size_t)N_NODES * HID;     // N*16
    float* h2   = h1   + (size_t)N_NODES * HID;     // N*16 (cols 10..15 unused)

    const int zn = N_NODES * (1 + 2 * HID);
    zero_kernel   <<<(zn + 255) / 256, 256, 0, stream>>>(deg, zn);
    degree_kernel <<<(N_EDGES + 255) / 256, 256, 0, stream>>>(dstI, deg);
    dinv_kernel   <<<(N_NODES + 255) / 256, 256, 0, stream>>>(deg);

    gemm1_kernel        <<<(TILES + 3) / 4, 128, 0, stream>>>(x, W1, h1);
    scatter_kernel<HID> <<<(N_EDGES + 255) / 256, 256, 0, stream>>>(srcI, dstI, deg, h1, agg1);
    post1_kernel        <<<(N_NODES * HID + 255) / 256, 256, 0, stream>>>(agg1, h1, deg, b1);

    gemm2_kernel        <<<(TILES + 3) / 4, 128, 0, stream>>>(agg1, W2, h2);
    scatter_kernel<NCLS><<<(N_EDGES + 255) / 256, 256, 0, stream>>>(srcI, dstI, deg, h2, agg2);
    softmax_kernel      <<<(N_NODES + 255) / 256, 256, 0, stream>>>(agg2, h2, deg, b2, (float*)d_out);
}